// MyGENConv_81381040325089
// MI455X (gfx1250) — compile-verified
//
#include <hip/hip_runtime.h>
#include <math.h>

#define NNODES   100000
#define NEDGES   1600000
#define IN_CH    128
#define OUT_CH   64
#define EDGE_CH  64
#define HID      128      // 2*OUT
#define EPS_MSG  1e-7f
#define EPS_SM   1e-16f
#define BN_EPS   1e-5f

#define NODE_TILES ((NNODES + 15) / 16)   // 6250 (exact)
#define EDGE_TILES (NEDGES / 16)          // 100000 (exact)

typedef float v2f __attribute__((ext_vector_type(2)));
typedef float v8f __attribute__((ext_vector_type(8)));

__device__ __forceinline__ v8f wmma_f32x4(v2f a, v2f b, v8f c) {
    // D = A(16x4 f32) x B(4x16 f32) + C(16x16 f32)
    return __builtin_amdgcn_wmma_f32_16x16x4_f32(
        /*neg_a=*/false, a, /*neg_b=*/false, b,
        /*c_mod=*/(short)0, c, /*reuse_a=*/false, /*reuse_b=*/false);
}

// ---------------------------------------------------------------------------
// K0: reset accumulation buffers (must run every launch: atomics accumulate)
// ---------------------------------------------------------------------------
__global__ void k_init(float* __restrict__ nmax, float* __restrict__ nsum,
                       float* __restrict__ agg, float* __restrict__ stats) {
    size_t idx = (size_t)blockIdx.x * blockDim.x + threadIdx.x;
    size_t n = (size_t)NNODES * OUT_CH;
    if (idx < n) {
        nmax[idx] = 0.0f;      // all msgs >= 1e-7 > 0, so 0 acts as -inf
        nsum[idx] = EPS_SM;    // fold "+eps" into the init
        agg[idx]  = 0.0f;
    }
    if (idx < 512) stats[idx] = 0.0f;   // colSum[128], colSq[128], scale, shift
}

// ---------------------------------------------------------------------------
// K1: x_src = x @ Wsrc^T, x_dst = x @ Wdst^T   [N,128]->[N,64] (both fused)
// one wave: 16-row tile, all 4 col-blocks, both weights (A reused 8x)
// ---------------------------------------------------------------------------
__global__ __launch_bounds__(128) void k_node_proj(
    const float* __restrict__ x, const float* __restrict__ wsrc,
    const float* __restrict__ wdst, float* __restrict__ xsrc,
    float* __restrict__ xdst) {
    __shared__ float lws[OUT_CH * IN_CH];   // 32 KB
    __shared__ float lwd[OUT_CH * IN_CH];   // 32 KB
    for (int i = threadIdx.x; i < OUT_CH * IN_CH; i += 128) {
        lws[i] = wsrc[i];
        lwd[i] = wdst[i];
    }
    __syncthreads();

    int wave = threadIdx.x >> 5, lane = threadIdx.x & 31;
    int rowTile = blockIdx.x * 4 + wave;
    if (rowTile >= NODE_TILES) return;          // wave-uniform
    int row0 = rowTile * 16;
    int half = lane >> 4, m = lane & 15;

    v8f accS[4] = {}, accD[4] = {};
    const float* xrow = x + (size_t)(row0 + m) * IN_CH;
    for (int k0 = 0; k0 < IN_CH; k0 += 4) {
        int kk = k0 + 2 * half;
        v2f a = *(const v2f*)(xrow + kk);
#pragma unroll
        for (int cb = 0; cb < 4; ++cb) {
            int n = cb * 16 + m;
            v2f bs = *(const v2f*)(&lws[n * IN_CH + kk]);
            v2f bd = *(const v2f*)(&lwd[n * IN_CH + kk]);
            accS[cb] = wmma_f32x4(a, bs, accS[cb]);
            accD[cb] = wmma_f32x4(a, bd, accD[cb]);
        }
    }
#pragma unroll
    for (int cb = 0; cb < 4; ++cb)
#pragma unroll
        for (int v = 0; v < 8; ++v) {
            size_t r = (size_t)(row0 + v + 8 * half);
            int c = cb * 16 + m;
            xsrc[r * OUT_CH + c] = accS[cb][v];
            xdst[r * OUT_CH + c] = accD[cb][v];
        }
}

// ---------------------------------------------------------------------------
// K2: msg = relu(x_src[src] + edge_attr @ We^T) + eps ; scatter segment-max
// one wave: 16-edge tile, all 64 output cols. atomicMax on int bits (msg>0)
// ---------------------------------------------------------------------------
__global__ __launch_bounds__(128) void k_edge_msg(
    const float* __restrict__ ea, const float* __restrict__ we,
    const int* __restrict__ src, const int* __restrict__ dst,
    const float* __restrict__ xsrc, float* __restrict__ msg,
    float* __restrict__ nmax) {
    __shared__ float lw[OUT_CH * EDGE_CH];  // 16 KB
    for (int i = threadIdx.x; i < OUT_CH * EDGE_CH; i += 128) lw[i] = we[i];
    __syncthreads();

    int wave = threadIdx.x >> 5, lane = threadIdx.x & 31;
    int tile = blockIdx.x * 4 + wave;       // EDGE_TILES is multiple of 4
    int e0 = tile * 16;
    int half = lane >> 4, m = lane & 15;

    v8f acc[4] = {};
    const float* arow = ea + (size_t)(e0 + m) * EDGE_CH;
    for (int k0 = 0; k0 < EDGE_CH; k0 += 4) {
        int kk = k0 + 2 * half;
        v2f a = *(const v2f*)(arow + kk);
#pragma unroll
        for (int cb = 0; cb < 4; ++cb) {
            v2f b = *(const v2f*)(&lw[(cb * 16 + m) * EDGE_CH + kk]);
            acc[cb] = wmma_f32x4(a, b, acc[cb]);
        }
    }
#pragma unroll
    for (int v = 0; v < 8; ++v) {
        int e = e0 + v + 8 * half;
        int s = src[e], d = dst[e];
        const float* xs = xsrc + (size_t)s * OUT_CH;
        size_t mrow = (size_t)e * OUT_CH;
        size_t drow = (size_t)d * OUT_CH;
#pragma unroll
        for (int cb = 0; cb < 4; ++cb) {
            int c = cb * 16 + m;
            float val = acc[cb][v] + xs[c];
            val = fmaxf(val, 0.0f) + EPS_MSG;
            msg[mrow + c] = val;
            atomicMax((int*)nmax + drow + c, __float_as_int(val));
        }
    }
}

// ---------------------------------------------------------------------------
// K3: segment-softmax denominator: nsum[dst] += exp(msg - nmax[dst])
// ---------------------------------------------------------------------------
__global__ __launch_bounds__(256) void k_soft_sum(
    const float* __restrict__ msg, const int* __restrict__ dst,
    const float* __restrict__ nmax, float* __restrict__ nsum) {
    size_t idx = (size_t)blockIdx.x * 256 + threadIdx.x;
    if (idx >= (size_t)NEDGES * OUT_CH) return;
    int e = (int)(idx >> 6), c = (int)(idx & 63);
    size_t drow = (size_t)dst[e] * OUT_CH + c;
    float w = expf(msg[idx] - nmax[drow]);
    atomicAdd(&nsum[drow], w);
}

// ---------------------------------------------------------------------------
// K4: agg[dst] += msg * exp(msg - nmax[dst]) / nsum[dst]
// ---------------------------------------------------------------------------
__global__ __launch_bounds__(256) void k_soft_agg(
    const float* __restrict__ msg, const int* __restrict__ dst,
    const float* __restrict__ nmax, const float* __restrict__ nsum,
    float* __restrict__ agg) {
    size_t idx = (size_t)blockIdx.x * 256 + threadIdx.x;
    if (idx >= (size_t)NEDGES * OUT_CH) return;
    int e = (int)(idx >> 6), c = (int)(idx & 63);
    size_t drow = (size_t)dst[e] * OUT_CH + c;
    float v = msg[idx];
    float alpha = expf(v - nmax[drow]) / nsum[drow];
    atomicAdd(&agg[drow], v * alpha);
}

// ---------------------------------------------------------------------------
// K5: h = (agg + x_dst) @ W1^T   [N,64]->[N,128]; A-fragment fuses the add
// ---------------------------------------------------------------------------
__global__ __launch_bounds__(128) void k_mlp1(
    const float* __restrict__ agg, const float* __restrict__ xdst,
    const float* __restrict__ w1, float* __restrict__ h) {
    __shared__ float lw[HID * OUT_CH];      // 32 KB
    for (int i = threadIdx.x; i < HID * OUT_CH; i += 128) lw[i] = w1[i];
    __syncthreads();

    int wave = threadIdx.x >> 5, lane = threadIdx.x & 31;
    int rowTile = blockIdx.x * 4 + wave;
    if (rowTile >= NODE_TILES) return;
    int row0 = rowTile * 16;
    int half = lane >> 4, m = lane & 15;

    v8f acc[8] = {};
    const float* ar = agg + (size_t)(row0 + m) * OUT_CH;
    const float* dr = xdst + (size_t)(row0 + m) * OUT_CH;
    for (int k0 = 0; k0 < OUT_CH; k0 += 4) {
        int kk = k0 + 2 * half;
        v2f a = *(const v2f*)(ar + kk) + *(const v2f*)(dr + kk);
#pragma unroll
        for (int cb = 0; cb < 8; ++cb) {
            v2f b = *(const v2f*)(&lw[(cb * 16 + m) * OUT_CH + kk]);
            acc[cb] = wmma_f32x4(a, b, acc[cb]);
        }
    }
#pragma unroll
    for (int cb = 0; cb < 8; ++cb)
#pragma unroll
        for (int v = 0; v < 8; ++v)
            h[(size_t)(row0 + v + 8 * half) * HID + cb * 16 + m] = acc[cb][v];
}

// ---------------------------------------------------------------------------
// K5b: per-column sum / sumsq via register accumulation + few atomics
// ---------------------------------------------------------------------------
__global__ __launch_bounds__(128) void k_stats(
    const float* __restrict__ h, float* __restrict__ colSum,
    float* __restrict__ colSq) {
    int c = threadIdx.x;                    // one column per thread, 128 cols
    float s = 0.0f, q = 0.0f;
    for (int r = blockIdx.x; r < NNODES; r += gridDim.x) {
        float v = h[(size_t)r * HID + c];
        s += v;
        q += v * v;
    }
    atomicAdd(&colSum[c], s);
    atomicAdd(&colSq[c], q);
}

// ---------------------------------------------------------------------------
// K6: fold BatchNorm into affine scale/shift
// ---------------------------------------------------------------------------
__global__ __launch_bounds__(128) void k_bn(
    const float* __restrict__ colSum, const float* __restrict__ colSq,
    const float* __restrict__ gamma, const float* __restrict__ beta,
    float* __restrict__ scale, float* __restrict__ shift) {
    int c = threadIdx.x;
    const float invn = 1.0f / (float)NNODES;
    float mean = colSum[c] * invn;
    float var = colSq[c] * invn - mean * mean;
    float sc = gamma[c] * rsqrtf(var + BN_EPS);
    scale[c] = sc;
    shift[c] = beta[c] - mean * sc;
}

// ---------------------------------------------------------------------------
// K7: out = relu(h*scale+shift) @ W2^T  [N,128]->[N,64]; BN+ReLU fused on A
// ---------------------------------------------------------------------------
__global__ __launch_bounds__(128) void k_mlp2(
    const float* __restrict__ h, const float* __restrict__ scale,
    const float* __restrict__ shift, const float* __restrict__ w2,
    float* __restrict__ out) {
    __shared__ float lw[OUT_CH * HID];      // 32 KB
    __shared__ float lsc[HID], lsh[HID];
    for (int i = threadIdx.x; i < OUT_CH * HID; i += 128) lw[i] = w2[i];
    if (threadIdx.x < HID) {
        lsc[threadIdx.x] = scale[threadIdx.x];
        lsh[threadIdx.x] = shift[threadIdx.x];
    }
    __syncthreads();

    int wave = threadIdx.x >> 5, lane = threadIdx.x & 31;
    int rowTile = blockIdx.x * 4 + wave;
    if (rowTile >= NODE_TILES) return;
    int row0 = rowTile * 16;
    int half = lane >> 4, m = lane & 15;

    v8f acc[4] = {};
    const float* hr = h + (size_t)(row0 + m) * HID;
    for (int k0 = 0; k0 < HID; k0 += 4) {
        int kk = k0 + 2 * half;
        v2f a;
        a.x = fmaxf(hr[kk] * lsc[kk] + lsh[kk], 0.0f);
        a.y = fmaxf(hr[kk + 1] * lsc[kk + 1] + lsh[kk + 1], 0.0f);
#pragma unroll
        for (int cb = 0; cb < 4; ++cb) {
            v2f b = *(const v2f*)(&lw[(cb * 16 + m) * HID + kk]);
            acc[cb] = wmma_f32x4(a, b, acc[cb]);
        }
    }
#pragma unroll
    for (int cb = 0; cb < 4; ++cb)
#pragma unroll
        for (int v = 0; v < 8; ++v)
            out[(size_t)(row0 + v + 8 * half) * OUT_CH + cb * 16 + m] =
                acc[cb][v];
}

// ---------------------------------------------------------------------------
extern "C" void kernel_launch(void* const* d_in, const int* in_sizes, int n_in,
                              void* d_out, int out_size, void* d_ws,
                              size_t ws_size, hipStream_t stream) {
    const float* x      = (const float*)d_in[0];
    const int*   eidx   = (const int*)d_in[1];
    const float* eattr  = (const float*)d_in[2];
    const float* wsrc   = (const float*)d_in[3];
    const float* wdst   = (const float*)d_in[4];
    const float* wedge  = (const float*)d_in[5];
    const float* w1     = (const float*)d_in[6];
    const float* gamma  = (const float*)d_in[7];
    const float* beta   = (const float*)d_in[8];
    const float* w2     = (const float*)d_in[9];
    float* out = (float*)d_out;

    const int* src = eidx;
    const int* dst = eidx + NEDGES;

    const size_t NOUT = (size_t)NNODES * OUT_CH;   // 6.4e6 floats
    float* ws    = (float*)d_ws;
    float* xsrc  = ws;
    float* xdst  = xsrc + NOUT;
    float* nmax  = xdst + NOUT;
    float* nsum  = nmax + NOUT;
    float* agg   = nsum + NOUT;
    float* h     = agg + NOUT;                     // N*128
    float* stats = h + (size_t)NNODES * HID;       // 512 floats
    float* colSum = stats;
    float* colSq  = stats + 128;
    float* scale  = stats + 256;
    float* shift  = stats + 384;
    float* msg   = stats + 512;                    // E*64 floats (410 MB)

    // K0: reset accumulators
    {
        int blocks = (int)((NOUT + 255) / 256);
        k_init<<<blocks, 256, 0, stream>>>(nmax, nsum, agg, stats);
    }
    // K1: node projections (src+dst fused)
    k_node_proj<<<(NODE_TILES + 3) / 4, 128, 0, stream>>>(x, wsrc, wdst, xsrc,
                                                          xdst);
    // K2: edge messages + segment max
    k_edge_msg<<<EDGE_TILES / 4, 128, 0, stream>>>(eattr, wedge, src, dst,
                                                   xsrc, msg, nmax);
    // K3: softmax denominators
    {
        size_t total = (size_t)NEDGES * OUT_CH;
        int blocks = (int)((total + 255) / 256);
        k_soft_sum<<<blocks, 256, 0, stream>>>(msg, dst, nmax, nsum);
        // K4: weighted aggregation
        k_soft_agg<<<blocks, 256, 0, stream>>>(msg, dst, nmax, nsum, agg);
    }
    // K5: MLP layer 1 (fused residual add)
    k_mlp1<<<(NODE_TILES + 3) / 4, 128, 0, stream>>>(agg, xdst, w1, h);
    // K5b: batch statistics
    k_stats<<<512, 128, 0, stream>>>(h, colSum, colSq);
    // K6: BN fold
    k_bn<<<1, 128, 0, stream>>>(colSum, colSq, gamma, beta, scale, shift);
    // K7: MLP layer 2 (fused BN + ReLU)
    k_mlp2<<<(NODE_TILES + 3) / 4, 128, 0, stream>>>(h, scale, shift, w2, out);
}